// cca_loss_44006234914852
// MI455X (gfx1250) — compile-verified
//
#include <hip/hip_runtime.h>
#include <math.h>

// ---------------- problem constants ----------------
static constexpr int Dd = 8192;   // feature dim
static constexpr int Kk = 10;     // outdim
static constexpr int Kp = 16;     // padded outdim for WMMA N-tile
static constexpr int Nn = 100;    // batch
static constexpr int Jj = 10;     // chunks = Nn/Kk
static constexpr float LRv  = 0.1f;
static constexpr float LR1v = 0.1f;

typedef float v2f __attribute__((ext_vector_type(2)));
typedef float v8f __attribute__((ext_vector_type(8)));

// ---------------- K1: column means of H1,H2 ----------------
__global__ void colmean_k(const float* __restrict__ H1, const float* __restrict__ H2,
                          float* __restrict__ m1, float* __restrict__ m2) {
  int i = blockIdx.x * blockDim.x + threadIdx.x;   // 0..2*Dd-1
  if (i >= 2 * Dd) return;
  const float* H = (i < Dd) ? H1 : H2;
  float* M       = (i < Dd) ? m1 : m2;
  int c          = (i < Dd) ? i : i - Dd;
  float s = 0.f;
  for (int n = 0; n < Nn; ++n) s += H[n * Dd + c];
  M[c] = s * (1.0f / Nn);
}

// ---------------- generic: out[d,:] = alpha*A[d,:]@M + beta*Cin[d,:] ----------------
__global__ void rowsmall_k(const float* __restrict__ A, const float* __restrict__ M,
                           const float* __restrict__ Cin, float* __restrict__ out,
                           float alpha, float beta) {
  int di = blockIdx.x * blockDim.x + threadIdx.x;
  if (di >= Dd) return;
  float a[Kk];
#pragma unroll
  for (int i = 0; i < Kk; ++i) a[i] = A[di * Kk + i];
#pragma unroll
  for (int c = 0; c < Kk; ++c) {
    float s = 0.f;
#pragma unroll
    for (int i = 0; i < Kk; ++i) s += a[i] * M[i * Kk + c];
    float r = alpha * s;
    if (Cin) r += beta * Cin[di * Kk + c];
    out[di * Kk + c] = r;
  }
}

// ---------------- reduction over d: o0 = x@P0, o1 = x@P1  (x = H - mean) ----------------
__global__ void rowdot2_k(const float* __restrict__ H, const float* __restrict__ mean,
                          const float* __restrict__ P0, const float* __restrict__ P1,
                          float* __restrict__ o0, float* __restrict__ o1) {
  int n = blockIdx.x;
  float acc[2 * Kk];
#pragma unroll
  for (int c = 0; c < 2 * Kk; ++c) acc[c] = 0.f;
  for (int d = threadIdx.x; d < Dd; d += 256) {
    float xv = H[n * Dd + d] - mean[d];
#pragma unroll
    for (int c = 0; c < Kk; ++c) acc[c] += xv * P0[d * Kk + c];
#pragma unroll
    for (int c = 0; c < Kk; ++c) acc[Kk + c] += xv * P1[d * Kk + c];
  }
  __shared__ float lds[2 * Kk][256];
#pragma unroll
  for (int c = 0; c < 2 * Kk; ++c) lds[c][threadIdx.x] = acc[c];
  __syncthreads();
  for (int st = 128; st; st >>= 1) {
    if (threadIdx.x < (unsigned)st)
#pragma unroll
      for (int c = 0; c < 2 * Kk; ++c) lds[c][threadIdx.x] += lds[c][threadIdx.x + st];
    __syncthreads();
  }
  if (threadIdx.x == 0)
#pragma unroll
    for (int c = 0; c < Kk; ++c) { o0[n * Kk + c] = lds[c][0]; o1[n * Kk + c] = lds[Kk + c][0]; }
}

// ---------------- 20 x (10x10 Gauss-Jordan inverse) -> assemble W ----------------
__global__ void ginv_k(const float* __restrict__ xQx, const float* __restrict__ yQy,
                       float* __restrict__ Wx, float* __restrict__ Wy) {
  int t = threadIdx.x;
  if (t >= 2 * Jj) return;
  const float* S = (t < Jj) ? xQx : yQy;
  float* W       = (t < Jj) ? Wx  : Wy;
  int j          = (t < Jj) ? t : t - Jj;
  float M[Kk][2 * Kk];
  for (int a = 0; a < Kk; ++a)
    for (int m = 0; m < Kk; ++m) {
      M[a][m]      = S[(m * Jj + j) * Kk + a];         // G_j[a][m] = xQx[m*J+j, a]
      M[a][Kk + m] = (a == m) ? 1.f : 0.f;
    }
  for (int col = 0; col < Kk; ++col) {
    int piv = col; float best = fabsf(M[col][col]);
    for (int r = col + 1; r < Kk; ++r) { float v = fabsf(M[r][col]); if (v > best) { best = v; piv = r; } }
    if (piv != col)
      for (int cc = 0; cc < 2 * Kk; ++cc) { float tmp = M[col][cc]; M[col][cc] = M[piv][cc]; M[piv][cc] = tmp; }
    float inv = 1.0f / M[col][col];
    for (int cc = 0; cc < 2 * Kk; ++cc) M[col][cc] *= inv;
    for (int r = 0; r < Kk; ++r)
      if (r != col) {
        float f = M[r][col];
        for (int cc = 0; cc < 2 * Kk; ++cc) M[r][cc] -= f * M[col][cc];
      }
  }
  for (int m = 0; m < Kk; ++m)
    for (int c = 0; c < Kk; ++c) W[(m * Jj + j) * Kk + c] = M[m][Kk + c];
}

// ---------------- Bp[n][0:16] = pad16( -(P @ T^T) - W/J )  (both sides) ----------------
__global__ void buildB_k(const float* __restrict__ yQT, const float* __restrict__ Tx,
                         const float* __restrict__ Wx, float* __restrict__ Bxp,
                         const float* __restrict__ xQT, const float* __restrict__ Ty,
                         const float* __restrict__ Wy, float* __restrict__ Byp) {
  int i = blockIdx.x * blockDim.x + threadIdx.x;
  if (i >= 2 * Nn * Kp) return;
  bool xs = i < Nn * Kp;
  int idx = xs ? i : i - Nn * Kp;
  int n = idx / Kp, c = idx % Kp;
  float r = 0.f;
  if (c < Kk) {
    const float* P = xs ? yQT : xQT;
    const float* T = xs ? Tx : Ty;
    const float* W = xs ? Wx : Wy;
    float s = 0.f;
#pragma unroll
    for (int a = 0; a < Kk; ++a) s += P[n * Kk + a] * T[c * Kk + a];
    r = -s - (1.0f / Jj) * W[n * Kk + c];
  }
  (xs ? Bxp : Byp)[idx] = r;
}

// ---------------- WMMA f32 GEMM: Vout = x^T[Dd x Nn] @ B[Nn x Kk] + Qadd ----------------
// B pre-padded to [Nn][16] and staged in LDS -> branch-free inner loop:
// 2x global_load_b32 (A) + 2x ds_load_b32 (B) + v_wmma_f32_16x16x4_f32.
__global__ void wmma_atb_k(const float* __restrict__ H, const float* __restrict__ mean,
                           const float* __restrict__ Bp, const float* __restrict__ Qadd,
                           float* __restrict__ Vout) {
  __shared__ float Bl[Nn * Kp];                       // 6.4 KB, shared by 8 waves
  for (int i = threadIdx.x; i < Nn * Kp; i += 256) Bl[i] = Bp[i];
  __syncthreads();

  int wave = (blockIdx.x * blockDim.x + threadIdx.x) >> 5;
  int lane = threadIdx.x & 31;
  int m0   = wave * 16;
  int hi   = lane >> 4;        // lane half selects K=0,1 vs K=2,3 in A/B VGPR layout
  int l16  = lane & 15;
  int m    = m0 + l16;
  float mn = mean[m];
  v8f acc = {};
  for (int k0 = 0; k0 < Nn; k0 += 4) {
    int ka = k0 + 2 * hi;
    v2f a, b;
    a.x = H[(ka + 0) * Dd + m] - mn;                 // A(m, ka)   = x[ka, m]
    a.y = H[(ka + 1) * Dd + m] - mn;                 // A(m, ka+1)
    b.x = Bl[(ka + 0) * Kp + l16];                   // B(ka,   n=l16), zero-padded cols
    b.y = Bl[(ka + 1) * Kp + l16];                   // B(ka+1, n=l16)
    acc = __builtin_amdgcn_wmma_f32_16x16x4_f32(false, a, false, b, (short)0, acc, false, false);
  }
  if (l16 < Kk) {
#pragma unroll
    for (int r = 0; r < 8; ++r) {                    // C/D: vgpr r -> row r + 8*hi, col l16
      int mm = m0 + r + 8 * hi;
      Vout[mm * Kk + l16] = acc[r] + Qadd[mm * Kk + l16];
    }
  }
}

// ---------------- C[a,b] = sum_d A[d,a]*B[d,b] (one block per entry) ----------------
__global__ void coldot_k(const float* __restrict__ A, const float* __restrict__ B,
                         float* __restrict__ C) {
  int a = blockIdx.x / Kk, b = blockIdx.x % Kk;
  __shared__ float lds[256];
  float s = 0.f;
  for (int d = threadIdx.x; d < Dd; d += 256) s += A[d * Kk + a] * B[d * Kk + b];
  lds[threadIdx.x] = s; __syncthreads();
  for (int st = 128; st; st >>= 1) {
    if (threadIdx.x < (unsigned)st) lds[threadIdx.x] += lds[threadIdx.x + st];
    __syncthreads();
  }
  if (threadIdx.x == 0) C[a * Kk + b] = lds[0];
}

// ---------------- per-column L2 norms of [Dd,Kk] ----------------
__global__ void colnorm_k(const float* __restrict__ A, float* __restrict__ nrm) {
  int c = blockIdx.x;
  __shared__ float lds[256];
  float s = 0.f;
  for (int d = threadIdx.x; d < Dd; d += 256) { float v = A[d * Kk + c]; s += v * v; }
  lds[threadIdx.x] = s; __syncthreads();
  for (int st = 128; st; st >>= 1) {
    if (threadIdx.x < (unsigned)st) lds[threadIdx.x] += lds[threadIdx.x + st];
    __syncthreads();
  }
  if (threadIdx.x == 0) nrm[c] = sqrtf(lds[0]);
}

// ---------------- A = Q - LR1 * V/norm (retraction direction, normalized) ----------------
__global__ void retrdir_k(const float* __restrict__ Q, const float* __restrict__ V,
                          const float* __restrict__ nrm, float* __restrict__ Aout) {
  int i = blockIdx.x * blockDim.x + threadIdx.x;
  if (i >= Dd * Kk) return;
  int c = i % Kk;
  Aout[i] = Q[i] - LR1v * V[i] / nrm[c];
}

// ---------------- CholeskyQR helper: from G = A^T A produce R^{-1} (upper) ----------------
__device__ void cholqr_rinv(const float* G, float* Rinv) {
  float L[Kk][Kk];
  for (int i = 0; i < Kk; ++i) for (int j = 0; j < Kk; ++j) L[i][j] = 0.f;
  for (int j = 0; j < Kk; ++j) {
    float s = G[j * Kk + j];
    for (int p = 0; p < j; ++p) s -= L[j][p] * L[j][p];
    float dj = sqrtf(fmaxf(s, 1e-30f));
    L[j][j] = dj;
    for (int i = j + 1; i < Kk; ++i) {
      float v = G[i * Kk + j];
      for (int p = 0; p < j; ++p) v -= L[i][p] * L[j][p];
      L[i][j] = v / dj;
    }
  }
  float Ri[Kk][Kk];                 // inverse of upper R = L^T  (R[i][p] = L[p][i])
  for (int i = 0; i < Kk; ++i) for (int j = 0; j < Kk; ++j) Ri[i][j] = 0.f;
  for (int j = 0; j < Kk; ++j) {
    Ri[j][j] = 1.0f / L[j][j];
    for (int i = j - 1; i >= 0; --i) {
      float s = 0.f;
      for (int p = i + 1; p <= j; ++p) s += L[p][i] * Ri[p][j];
      Ri[i][j] = -s / L[i][i];
    }
  }
  for (int i = 0; i < Kk; ++i) for (int j = 0; j < Kk; ++j) Rinv[i * Kk + j] = Ri[i][j];
}

__global__ void cholinv_k(const float* __restrict__ Gx, const float* __restrict__ Gy,
                          float* __restrict__ Rix, float* __restrict__ Riy) {
  if (threadIdx.x == 0) cholqr_rinv(Gx, Rix);
  else if (threadIdx.x == 1) cholqr_rinv(Gy, Riy);
}

// ---------------- rotation-space update: T{x,y}_new (all 10x10 math in one block) ----------------
__global__ void tnew_k(const float* __restrict__ xQx, const float* __restrict__ yQT,
                       const float* __restrict__ yQy, const float* __restrict__ xQT,
                       const float* __restrict__ Tx, const float* __restrict__ Ty,
                       float* __restrict__ Txn, float* __restrict__ Tyn) {
  __shared__ float Vt[2][Kk][Kk];
  __shared__ float Mm[2][Kk][Kk];
  __shared__ float nrm[2][Kk];
  int t = threadIdx.x;
  if (t < 200) {                                    // VT = -(proj)^T @ (otherQT)
    int side = t / 100, e = t % 100, a = e / Kk, b = e % Kk;
    const float* P = side ? yQy : xQx;
    const float* Q = side ? xQT : yQT;
    float s = 0.f;
    for (int n = 0; n < Nn; ++n) s += P[n * Kk + a] * Q[n * Kk + b];
    Vt[side][a][b] = -s;
  }
  __syncthreads();
  if (t < 200) {                                    // antisymmetrize
    int side = t / 100, e = t % 100, a = e / Kk, b = e % Kk;
    Mm[side][a][b] = Vt[side][a][b] - Vt[side][b][a];
  }
  __syncthreads();
  if (t < 2 * Kk) {                                 // column norms
    int side = t / Kk, c = t % Kk; float s = 0.f;
    for (int a = 0; a < Kk; ++a) { float v = Mm[side][a][c]; s += v * v; }
    nrm[side][c] = sqrtf(s);
  }
  __syncthreads();
  if (t < 200) {                                    // Bmat = I - LR * 0.5*M/norm
    int side = t / 100, e = t % 100, a = e / Kk, b = e % Kk;
    float v = 0.5f * Mm[side][a][b] / nrm[side][b];
    Vt[side][a][b] = ((a == b) ? 1.f : 0.f) - LRv * v;
  }
  __syncthreads();
  if (t < 2) {                                      // retrqr(I, -LR*VT) via CholQR; Tnew = T@Qbar
    float G[Kk * Kk];
    for (int a = 0; a < Kk; ++a)
      for (int b = 0; b < Kk; ++b) {
        float s = 0.f;
        for (int p = 0; p < Kk; ++p) s += Vt[t][p][a] * Vt[t][p][b];
        G[a * Kk + b] = s;
      }
    float Ri[Kk * Kk];
    cholqr_rinv(G, Ri);
    float Qb[Kk][Kk];
    for (int i = 0; i < Kk; ++i)
      for (int j = 0; j < Kk; ++j) {
        float s = 0.f;
        for (int p = 0; p < Kk; ++p) s += Vt[t][i][p] * Ri[p * Kk + j];
        Qb[i][j] = s;
      }
    const float* T = t ? Ty : Tx;
    float* Tn      = t ? Tyn : Txn;
    for (int i = 0; i < Kk; ++i)
      for (int j = 0; j < Kk; ++j) {
        float s = 0.f;
        for (int p = 0; p < Kk; ++p) s += T[i * Kk + p] * Qb[p][j];
        Tn[i * Kk + j] = s;
      }
  }
}

// ---------------- final: -sum_c (sum wx*wy)/sqrt(sum wx^2 * sum wy^2) ----------------
__global__ void corr_k(const float* __restrict__ wx, const float* __restrict__ wy,
                       float* __restrict__ out) {
  __shared__ float terms[Kk];
  int t = threadIdx.x;
  if (t < Kk) {
    float sxy = 0.f, sxx = 0.f, syy = 0.f;
    for (int n = 0; n < Nn; ++n) {
      float a = wx[n * Kk + t], b = wy[n * Kk + t];
      sxy += a * b; sxx += a * a; syy += b * b;
    }
    terms[t] = sxy / sqrtf(sxx * syy);
  }
  __syncthreads();
  if (t == 0) {
    float s = 0.f;
    for (int c = 0; c < Kk; ++c) s += terms[c];
    out[0] = -s;
  }
}

// =====================================================================
extern "C" void kernel_launch(void* const* d_in, const int* in_sizes, int n_in,
                              void* d_out, int out_size, void* d_ws, size_t ws_size,
                              hipStream_t stream) {
  (void)in_sizes; (void)n_in; (void)out_size; (void)ws_size;
  const float* H1 = (const float*)d_in[0];
  const float* H2 = (const float*)d_in[1];
  const float* Qx = (const float*)d_in[2];
  const float* Qy = (const float*)d_in[3];
  const float* Tx = (const float*)d_in[4];
  const float* Ty = (const float*)d_in[5];
  float* outp = (float*)d_out;

  // bump-allocate workspace (floats); total ~3.4 MB
  float* p = (float*)d_ws;
  auto alloc = [&](size_t n) { float* r = p; p += n; return r; };
  float* mean1 = alloc(Dd);        float* mean2 = alloc(Dd);
  float* QxTx  = alloc(Dd * Kk);   float* QyTy  = alloc(Dd * Kk);
  float* xQT   = alloc(Nn * Kk);   float* xQx   = alloc(Nn * Kk);
  float* yQT   = alloc(Nn * Kk);   float* yQy   = alloc(Nn * Kk);
  float* Wx    = alloc(Nn * Kk);   float* Wy    = alloc(Nn * Kk);
  float* Bxp   = alloc(Nn * Kp);   float* Byp   = alloc(Nn * Kp);
  float* Sx    = alloc(Kk * Kk);   float* Sy    = alloc(Kk * Kk);
  float* cnx   = alloc(Kk);        float* cny   = alloc(Kk);
  float* Gxc   = alloc(Kk * Kk);   float* Gyc   = alloc(Kk * Kk);
  float* Rix   = alloc(Kk * Kk);   float* Riy   = alloc(Kk * Kk);
  float* Txn   = alloc(Kk * Kk);   float* Tyn   = alloc(Kk * Kk);
  float* wx    = alloc(Nn * Kk);   float* wy    = alloc(Nn * Kk);
  float* dum   = alloc(Nn * Kk);
  float* Vx    = alloc(Dd * Kk);   float* Vy    = alloc(Dd * Kk);
  float* ArX   = alloc(Dd * Kk);   float* ArY   = alloc(Dd * Kk);
  float* QxN   = alloc(Dd * Kk);   float* QyN   = alloc(Dd * Kk);
  float* AfX   = alloc(Dd * Kk);   float* AfY   = alloc(Dd * Kk);

  // 1) centering means
  colmean_k<<<(2 * Dd + 255) / 256, 256, 0, stream>>>(H1, H2, mean1, mean2);
  // 2) QxTx = Qx@Tx, QyTy = Qy@Ty
  rowsmall_k<<<Dd / 256, 256, 0, stream>>>(Qx, Tx, nullptr, QxTx, 1.f, 0.f);
  rowsmall_k<<<Dd / 256, 256, 0, stream>>>(Qy, Ty, nullptr, QyTy, 1.f, 0.f);
  // 3) projections xQT=x@(QxTx), xQx=x@Qx (and y-side)
  rowdot2_k<<<Nn, 256, 0, stream>>>(H1, mean1, QxTx, Qx, xQT, xQx);
  rowdot2_k<<<Nn, 256, 0, stream>>>(H2, mean2, QyTy, Qy, yQT, yQy);
  // 4) 20 small inverses -> W (G_j is a row-slice of xQx/yQy)
  ginv_k<<<1, 32, 0, stream>>>(xQx, yQy, Wx, Wy);
  // 5) fused, 16-column-padded RHS for gradient GEMM
  buildB_k<<<(2 * Nn * Kp + 255) / 256, 256, 0, stream>>>(yQT, Tx, Wx, Bxp, xQT, Ty, Wy, Byp);
  // 6) Vx = x^T @ Bx + Qx  (WMMA f32 16x16x4, branch-free inner loop), same for Vy
  wmma_atb_k<<<(Dd / 16) / 8, 256, 0, stream>>>(H1, mean1, Bxp, Qx, Vx);
  wmma_atb_k<<<(Dd / 16) / 8, 256, 0, stream>>>(H2, mean2, Byp, Qy, Vy);
  // 7) tangent projection: V -= Q @ (V^T Q)
  coldot_k<<<Kk * Kk, 256, 0, stream>>>(Vx, Qx, Sx);
  coldot_k<<<Kk * Kk, 256, 0, stream>>>(Vy, Qy, Sy);
  rowsmall_k<<<Dd / 256, 256, 0, stream>>>(Qx, Sx, Vx, Vx, -1.f, 1.f);
  rowsmall_k<<<Dd / 256, 256, 0, stream>>>(Qy, Sy, Vy, Vy, -1.f, 1.f);
  // 8) column-normalize V, form A = Q - LR1*V/||V||
  colnorm_k<<<Kk, 256, 0, stream>>>(Vx, cnx);
  colnorm_k<<<Kk, 256, 0, stream>>>(Vy, cny);
  retrdir_k<<<(Dd * Kk + 255) / 256, 256, 0, stream>>>(Qx, Vx, cnx, ArX);
  retrdir_k<<<(Dd * Kk + 255) / 256, 256, 0, stream>>>(Qy, Vy, cny, ArY);
  // 9) CholeskyQR retraction: Qnew = A @ chol(A^T A)^{-T}
  coldot_k<<<Kk * Kk, 256, 0, stream>>>(ArX, ArX, Gxc);
  coldot_k<<<Kk * Kk, 256, 0, stream>>>(ArY, ArY, Gyc);
  cholinv_k<<<1, 32, 0, stream>>>(Gxc, Gyc, Rix, Riy);
  rowsmall_k<<<Dd / 256, 256, 0, stream>>>(ArX, Rix, nullptr, QxN, 1.f, 0.f);
  rowsmall_k<<<Dd / 256, 256, 0, stream>>>(ArY, Riy, nullptr, QyN, 1.f, 0.f);
  // 10) rotation update (all k x k math)
  tnew_k<<<1, 256, 0, stream>>>(xQx, yQT, yQy, xQT, Tx, Ty, Txn, Tyn);
  // 11) Afinal = Qnew @ Tnew; wx = x@Afinal, wy = y@Afinal
  rowsmall_k<<<Dd / 256, 256, 0, stream>>>(QxN, Txn, nullptr, AfX, 1.f, 0.f);
  rowsmall_k<<<Dd / 256, 256, 0, stream>>>(QyN, Tyn, nullptr, AfY, 1.f, 0.f);
  rowdot2_k<<<Nn, 256, 0, stream>>>(H1, mean1, AfX, AfX, wx, dum);
  rowdot2_k<<<Nn, 256, 0, stream>>>(H2, mean2, AfY, AfY, wy, dum);
  // 12) scalar objective (Du/column-scaling cancels analytically to this form)
  corr_k<<<1, 32, 0, stream>>>(wx, wy, outp);
}